// Split_CNN_5609227289126
// MI455X (gfx1250) — compile-verified
//
#include <hip/hip_runtime.h>
#include <hip/hip_bf16.h>

#define B_DIM 8192
#define W_DIM 64
#define C_DIM 128
#define D_DIM 192   // 3 * W
#define F_DIM 120
#define BN_EPS_F 0.001f

typedef __attribute__((ext_vector_type(2))) float v2f;
typedef __attribute__((ext_vector_type(8))) float v8f;
typedef __attribute__((ext_vector_type(4))) int   v4i;

typedef __attribute__((address_space(1))) v4i gv4i;   // global
typedef __attribute__((address_space(3))) v4i lv4i;   // LDS

#if defined(__has_builtin)
#  if __has_builtin(__builtin_amdgcn_global_load_async_to_lds_b128)
#    define HAVE_ASYNC_LDS 1
#  endif
#endif

// Branch-free tanh: 1 - 2/(exp(2x)+1).
// exp overflow/underflow gives exact +/-1 saturation; no EXEC divergence.
__device__ __forceinline__ float fast_tanh(float x) {
    float e = __expf(2.0f * x);                     // v_exp_f32 path
    return 1.0f - 2.0f * __builtin_amdgcn_rcpf(e + 1.0f);
}

// ---------------------------------------------------------------------------
// Kernel 1: fused center-tap conv + tanh + BN affine.  x_bn[b][d], d = w*3+f.
// One block per batch row; streams the 256MB input exactly once, coalesced,
// via async global->LDS b128 copies (ASYNCcnt) when available.
// ---------------------------------------------------------------------------
#define ROW_STRIDE 132   // mult of 4 (16B-aligned chunks); 132 mod 64 = 4 banks spread

__global__ __launch_bounds__(256) void conv_bn_kernel(
    const float* __restrict__ in,      // (B, W, C)
    const float* __restrict__ conv_w,  // (W, 3, C, 3)
    const float* __restrict__ conv_b,  // (W, 3)
    const float* __restrict__ gamma,   // (D)
    const float* __restrict__ beta,    // (D)
    const float* __restrict__ run_mean,// (D)
    const float* __restrict__ run_var, // (D)
    float* __restrict__ xout)          // (B, D)
{
    __shared__ __align__(16) float row[W_DIM * ROW_STRIDE];
    const int b = blockIdx.x;
    const float* inb = in + (size_t)b * (W_DIM * C_DIM);

#if HAVE_ASYNC_LDS
    // 2048 x b128 async copies: global -> LDS, no VGPR round trip.
    for (int i = threadIdx.x; i < (W_DIM * C_DIM) / 4; i += 256) {
        int w  = i >> 5;            // 32 float4 chunks per w
        int c4 = (i & 31) * 4;
        __builtin_amdgcn_global_load_async_to_lds_b128(
            (gv4i*)(inb + (size_t)i * 4),
            (lv4i*)&row[w * ROW_STRIDE + c4],
            0, 0);
    }
#  if __has_builtin(__builtin_amdgcn_s_wait_asynccnt)
    __builtin_amdgcn_s_wait_asynccnt(0);
#  else
    asm volatile("s_wait_asynccnt 0" ::: "memory");
#  endif
#else
    for (int i = threadIdx.x; i < (W_DIM * C_DIM) / 4; i += 256) {
        int w  = i >> 5;
        int c4 = (i & 31) * 4;
        float4 v = *(const float4*)(inb + (size_t)i * 4);
        *(float4*)&row[w * ROW_STRIDE + c4] = v;
    }
#endif
    __syncthreads();

    const int d = threadIdx.x;
    if (d < D_DIM) {
        const int w = d / 3;
        const int f = d - 3 * w;
        // wc = conv_w[w, 1, :, f]  (only the center tap is used)
        const float* wc = conv_w + ((size_t)(w * 3 + 1)) * (C_DIM * 3) + f;
        const float* rw = row + w * ROW_STRIDE;
        float acc = conv_b[d];
        #pragma unroll 8
        for (int c = 0; c < C_DIM; ++c)
            acc = fmaf(rw[c], wc[c * 3], acc);
        float v = fast_tanh(acc);
        float g = gamma[d] * rsqrtf(run_var[d] + BN_EPS_F);
        xout[(size_t)b * D_DIM + d] = (v - run_mean[d]) * g + beta[d];
    }
}

// ---------------------------------------------------------------------------
// Kernel 2: fused fc1(+tanh) and fc2 with fp32 WMMA (V_WMMA_F32_16X16X4_F32).
// 4 waves cooperate on one 16-row tile (shared xs/hs, ~24.5KB LDS):
//   fc1: 12 N-tiles -> 3 per wave;  fc2: 8 N-tiles (120 padded to 128) -> 2.
// A-frag (16x4 f32): lane<16 -> K={0,1}, lane>=16 -> K={2,3}; M = lane&15.
// C/D (16x16 f32):   VGPR j -> M = j + (lane>=16 ? 8 : 0), N = lane&15.
// ---------------------------------------------------------------------------
#define XS_STRIDE 196   // mult of 4; 196 mod 64 = 4 -> conflict-free frag reads

__global__ __launch_bounds__(128) void mlp_wmma_kernel(
    const float* __restrict__ x,    // (B, D)
    const float* __restrict__ w1,   // (D, D)
    const float* __restrict__ b1,   // (D)
    const float* __restrict__ w2,   // (D, F)
    const float* __restrict__ b2,   // (F)
    float* __restrict__ out)        // (B, F)
{
    __shared__ __align__(16) float xw[16 * XS_STRIDE];
    __shared__ __align__(16) float hw[16 * XS_STRIDE];

    const int wv   = threadIdx.x >> 5;
    const int lane = threadIdx.x & 31;
    const int row0 = blockIdx.x * 16;

    // stage 16x192 activation tile (b128, coalesced)
    for (int i = threadIdx.x; i < 16 * D_DIM / 4; i += 128) {
        int r  = i / 48;            // 48 float4 chunks per row
        int c4 = (i - r * 48) * 4;
        float4 v = *(const float4*)(x + (size_t)(row0 + r) * D_DIM + c4);
        *(float4*)&xw[r * XS_STRIDE + c4] = v;
    }
    __syncthreads();

    const int m    = lane & 15;
    const int koff = (lane >> 4) * 2;   // 0 for lanes 0-15, 2 for lanes 16-31
    const int hi8  = (lane >> 4) * 8;

    // ---- fc1: h = tanh(x @ w1 + b1); wave wv owns N-tiles {3wv..3wv+2} ----
    for (int t = 0; t < 3; ++t) {
        const int nt = wv * 3 + t;
        const int n  = nt * 16 + m;
        v8f acc = {};
        #pragma unroll 4
        for (int kt = 0; kt < D_DIM / 4; ++kt) {
            const int k = kt * 4 + koff;
            v2f a = *(const v2f*)(xw + m * XS_STRIDE + k);
            v2f bm;
            bm.x = w1[(size_t)k * D_DIM + n];
            bm.y = w1[(size_t)(k + 1) * D_DIM + n];
            acc = __builtin_amdgcn_wmma_f32_16x16x4_f32(
                false, a, false, bm, (short)0, acc, false, false);
        }
        const float bias = b1[n];
        #pragma unroll
        for (int j = 0; j < 8; ++j) {
            const int mm = j + hi8;
            hw[mm * XS_STRIDE + n] = fast_tanh(acc[j] + bias);
        }
    }
    __syncthreads();

    // ---- fc2: out = h @ w2 + b2; wave wv owns N-tiles {2wv, 2wv+1} ----
    for (int t = 0; t < 2; ++t) {
        const int nt = wv * 2 + t;
        const int n  = nt * 16 + m;
        const bool valid = (n < F_DIM);
        v8f acc = {};
        #pragma unroll 4
        for (int kt = 0; kt < D_DIM / 4; ++kt) {
            const int k = kt * 4 + koff;
            v2f a = *(const v2f*)(hw + m * XS_STRIDE + k);
            v2f bm;
            bm.x = valid ? w2[(size_t)k * F_DIM + n] : 0.0f;
            bm.y = valid ? w2[(size_t)(k + 1) * F_DIM + n] : 0.0f;
            acc = __builtin_amdgcn_wmma_f32_16x16x4_f32(
                false, a, false, bm, (short)0, acc, false, false);
        }
        if (valid) {
            const float bias = b2[n];
            #pragma unroll
            for (int j = 0; j < 8; ++j) {
                const int mm = j + hi8;
                out[(size_t)(row0 + mm) * F_DIM + n] = acc[j] + bias;
            }
        }
    }
}

// ---------------------------------------------------------------------------
extern "C" void kernel_launch(void* const* d_in, const int* in_sizes, int n_in,
                              void* d_out, int out_size, void* d_ws, size_t ws_size,
                              hipStream_t stream) {
    (void)in_sizes; (void)n_in; (void)out_size; (void)ws_size;

    const float* in       = (const float*)d_in[0];
    const float* conv_w   = (const float*)d_in[1];
    const float* conv_b   = (const float*)d_in[2];
    const float* gamma    = (const float*)d_in[3];
    const float* beta     = (const float*)d_in[4];
    const float* run_mean = (const float*)d_in[5];
    const float* run_var  = (const float*)d_in[6];
    const float* w1       = (const float*)d_in[7];
    const float* b1       = (const float*)d_in[8];
    const float* w2       = (const float*)d_in[9];
    const float* b2       = (const float*)d_in[10];

    float* xbn = (float*)d_ws;          // (B, D) fp32 = ~6.3 MB scratch
    float* out = (float*)d_out;         // (B, F)

    conv_bn_kernel<<<B_DIM, 256, 0, stream>>>(
        in, conv_w, conv_b, gamma, beta, run_mean, run_var, xbn);

    mlp_wmma_kernel<<<B_DIM / 16, 128, 0, stream>>>(
        xbn, w1, b1, w2, b2, out);
}